// YoloTarget_62947040690647
// MI455X (gfx1250) — compile-verified
//
#include <hip/hip_runtime.h>

// ---------------------------------------------------------------------------
// YOLO target assignment (B=64, T=60, K=3, S=80, C=85)
//   out[0 .. GT)     = ground_true (418 MB, mostly 0)
//   out[GT .. GT+NO) = no_obj_mask (4.9 MB, mostly 1)
//
// Store-bandwidth bound: ~423 MB of streaming writes. Roofline floor at
// 23.3 TB/s HBM ~= 18 us. No matmul anywhere -> WMMA inapplicable; the
// gfx1250 path that matters is wave32 + global_store_b128 th:TH_STORE_NT.
// Each thread issues 8 b128 NT stores so the per-wave scalar preamble
// (~15 instrs) amortizes over 8 VMEM ops instead of 1.
// ---------------------------------------------------------------------------

#define B_  64
#define T_  60
#define K_  3
#define S_  80
#define C_  85

static constexpr int GT_FLOATS = B_ * K_ * S_ * S_ * C_;   // 104,448,000
static constexpr int NO_FLOATS = B_ * K_ * S_ * S_;        //   1,228,800
static constexpr int GT4   = GT_FLOATS / 4;                //  26,112,000
static constexpr int TOT4  = (GT_FLOATS + NO_FLOATS) / 4;  //  26,419,200

static constexpr int FILL_THREADS = 256;
static constexpr int FILL_PER_THR = 8;                       // 8 x b128 per thread
static constexpr int FILL_TILE    = FILL_THREADS * FILL_PER_THR;        // 2048
static constexpr int FILL_BLOCKS  = TOT4 / FILL_TILE;                   // 12,900 exact
static constexpr int GT_BLOCKS    = GT4 / FILL_TILE;                    // 12,750 exact
// -> the 0.0 / 1.0 boundary falls on a block-tile boundary.

typedef float v4f __attribute__((ext_vector_type(4)));

// ---------------------------------------------------------------------------
// Kernel 1: stream-fill. Block tile = 2048 consecutive float4s; every block
// is entirely in the ground_true region (0.0f) or the no_obj region (1.0f),
// so the fill value is a per-block scalar. Each store instruction covers a
// contiguous 512 B per wave32.
// ---------------------------------------------------------------------------
__global__ void yolo_fill_kernel(v4f* __restrict__ out) {
    const int  base = blockIdx.x * FILL_TILE + threadIdx.x;
    const float f   = (blockIdx.x < GT_BLOCKS) ? 0.0f : 1.0f;
    const v4f  v    = { f, f, f, f };
#pragma unroll
    for (int k = 0; k < FILL_PER_THR; ++k) {
        __builtin_nontemporal_store(v, out + base + k * FILL_THREADS);
    }
}

// ---------------------------------------------------------------------------
// Kernel 2: per-target IoU argmax against 9 anchors (ANCHORS / STRIDE baked
// in), then scatter x,y,w,h,obj,class + clear no_obj. mode='drop' semantics:
// out-of-scale targets had j forced to 80 (out of range) in the reference,
// so every write for them drops -> we simply skip them.
// ---------------------------------------------------------------------------
__global__ void yolo_scatter_kernel(const float* __restrict__ targets_xywh,
                                    const int*   __restrict__ target_cls,
                                    const int*   __restrict__ valid,
                                    float*       __restrict__ out) {
    int t = blockIdx.x * blockDim.x + threadIdx.x;
    if (t >= B_ * T_) return;

    const int b = t / T_;

    const float x = targets_xywh[t * 4 + 0] * (float)S_;
    const float y = targets_xywh[t * 4 + 1] * (float)S_;
    const float w = targets_xywh[t * 4 + 2] * (float)S_;
    const float h = targets_xywh[t * 4 + 3] * (float)S_;

    // ANCHORS / STRIDE(=8)
    const float aw[9] = { 1.25f, 2.0f,   4.125f, 3.75f,  7.75f,
                          7.375f, 14.5f, 19.5f,  46.625f };
    const float ah[9] = { 1.625f, 3.75f, 2.875f, 7.625f, 5.625f,
                          14.875f, 11.25f, 24.75f, 40.75f };

    const float wh = w * h;
    int   best     = 0;
    float best_iou = -1.0f;
#pragma unroll
    for (int a = 0; a < 9; ++a) {
        float inter = fminf(w, aw[a]) * fminf(h, ah[a]);
        inter       = fmaxf(inter, 0.0f);
        float uni   = wh + aw[a] * ah[a] - inter;
        float iou   = inter / uni;
        if (iou > best_iou) { best_iou = iou; best = a; }   // first-max wins
    }

    // ANCHOR_MASK = {0,1,2}; k = best % 3 == best when in scale.
    const bool in_scale = (best < K_) && (valid[t] != 0);
    if (!in_scale) return;   // reference sets j=80 -> all writes dropped

    const int k = best;
    const int i = (int)floorf(x);
    const int j = (int)floorf(y);
    if ((unsigned)i >= (unsigned)S_ || (unsigned)j >= (unsigned)S_) return; // drop

    const long cell = (((long)(b * K_ + k) * S_ + j) * S_ + i);
    float* gt = out + cell * C_;

    gt[0] = x;
    gt[1] = y;
    gt[2] = w;
    gt[3] = h;
    gt[4] = 1.0f;
    const int c = target_cls[t] + 5;
    if (c >= 0 && c < C_) gt[c] = 1.0f;

    out[GT_FLOATS + cell] = 0.0f;   // no_obj_mask[b,k,j,i] = 0
}

// ---------------------------------------------------------------------------
extern "C" void kernel_launch(void* const* d_in, const int* in_sizes, int n_in,
                              void* d_out, int out_size, void* d_ws, size_t ws_size,
                              hipStream_t stream) {
    (void)in_sizes; (void)n_in; (void)out_size; (void)d_ws; (void)ws_size;

    const float* targets_xywh = (const float*)d_in[0];
    const int*   target_cls   = (const int*)d_in[1];
    const int*   valid        = (const int*)d_in[2];
    float*       out          = (float*)d_out;

    // Pass 1: bandwidth-bound fill (423 MB of NT b128 stores, 8 per thread).
    yolo_fill_kernel<<<FILL_BLOCKS, FILL_THREADS, 0, stream>>>((v4f*)out);

    // Pass 2: tiny scatter (3840 threads = 120 wave32).
    yolo_scatter_kernel<<<(B_ * T_ + 63) / 64, 64, 0, stream>>>(
        targets_xywh, target_cls, valid, out);
}